// Rasterize_52321291600271
// MI455X (gfx1250) — compile-verified
//
#include <hip/hip_runtime.h>
#include <hip/hip_bf16.h>

typedef __attribute__((ext_vector_type(2))) float v2f;
typedef __attribute__((ext_vector_type(8))) float v8f;

#define NB 4
#define NVERT 1024
#define NFACE 1024
#define RSIZE 256
#define NP 4            // pixel tiles (of 16) per wave
#define FTILES (NFACE / 16)

// ---------------------------------------------------------------------------
// Kernel 1: build edge-function coefficients (A,B,C,0) per (batch, edge, face)
// laid out as float4 coef[b][e][f] so the main kernel's A-fragment loads are
// 16B/face contiguous per 16-face tile.
// cross(px,py) = A*px + B*py + C,  A=-ey, B=ex, C=ey*vx - ex*vy
// ---------------------------------------------------------------------------
__global__ void rast_prep(const float* __restrict__ verts,
                          const int* __restrict__ faces,
                          float4* __restrict__ coef) {
    int i = blockIdx.x * blockDim.x + threadIdx.x;   // [0, NB*3*NFACE)
    if (i >= NB * 3 * NFACE) return;
    int f  = i & (NFACE - 1);
    int be = i >> 10;
    int e  = be % 3;
    int b  = be / 3;

    const int* fp = faces + ((size_t)b * NFACE + f) * 3;
    int i0 = fp[e];
    int i1 = fp[(e + 1) % 3];
    const float* vb = verts + (size_t)b * NVERT * 2;
    float vx = vb[i0 * 2 + 0], vy = vb[i0 * 2 + 1];
    float wx = vb[i1 * 2 + 0], wy = vb[i1 * 2 + 1];
    float ex = wx - vx, ey = wy - vy;
    float4 c;
    c.x = -ey;                 // A (coeff of px)
    c.y = ex;                  // B (coeff of py)
    c.z = ey * vx - ex * vy;   // C
    c.w = 0.0f;                // K=3 pad
    coef[i] = c;
}

// ---------------------------------------------------------------------------
// Kernel 2: WMMA rasterizer. One wave owns 64 consecutive raster pixels
// (4 x 16-pixel tiles on one row) and sweeps all 64 face tiles.
// Per face tile: 3x v_wmma_f32_16x16x4_f32 per pixel tile (edges 0/1/2),
// min3 per face (forward winding) / max3 per face (reverse winding), then an
// in-iteration v_max3/v_min3 reduction over the 8 rows held per lane, folding
// the loop-carried scalar accumulator into the last 3-input op.
// Pixel covered iff  max_f min3 > 0  OR  min_f max3 < 0.
// ---------------------------------------------------------------------------
__global__ void __launch_bounds__(256)
rast_main(const float* __restrict__ coef, float* __restrict__ raster) {
    const int lane = threadIdx.x & 31;
    const int w    = blockIdx.x * (blockDim.x >> 5) + (threadIdx.x >> 5); // wave id
    const int b    = w >> 10;          // batch
    const int g    = w & 1023;         // pixel-group within batch
    const int ry   = g >> 2;           // raster row
    const int rx0  = (g & 3) << 6;     // first raster col of the 64-pixel strip

    // ---- B-matrix fragments (4x16, K x N), one per pixel tile -------------
    // VGPR0: lanes 0-15 -> K=0 (px), lanes 16-31 -> K=2 (1.0)
    // VGPR1: lanes 0-15 -> K=1 (py), lanes 16-31 -> K=3 (0.0)
    const int  fl   = lane & 15;
    const bool lowh = lane < 16;
    const float py  = (2.0f * (float)ry + 1.0f - (float)RSIZE) * (1.0f / RSIZE);
    v2f bfrag[NP];
#pragma unroll
    for (int t = 0; t < NP; ++t) {
        int   rx = rx0 + t * 16 + fl;
        float px = (2.0f * (float)rx + 1.0f - (float)RSIZE) * (1.0f / RSIZE);
        bfrag[t].x = lowh ? px : 1.0f;
        bfrag[t].y = lowh ? py : 0.0f;
    }

    // ---- scalar (per-lane) accumulators -----------------------------------
    float accP[NP];   // running max over faces of min3(cross)  (fwd winding)
    float accN[NP];   // running min over faces of max3(cross)  (rev winding)
#pragma unroll
    for (int t = 0; t < NP; ++t) { accP[t] = -3.4e38f; accN[t] = 3.4e38f; }

    // ---- A-fragment lane addressing (16x4 f32 A layout) --------------------
    // lanes 0-15: floats {A,B} of face fl; lanes 16-31: floats {C,0}
    const float* cb = coef + (size_t)b * 3 * NFACE * 4
                    + (size_t)fl * 4 + (size_t)(lane >> 4) * 2;

#pragma unroll 2
    for (int ft = 0; ft < FTILES; ++ft) {
        const float* p0 = cb + (size_t)ft * 64;
        v2f a0 = *(const v2f*)(p0);                 // edge 0 coeffs
        v2f a1 = *(const v2f*)(p0 + NFACE * 4);     // edge 1
        v2f a2 = *(const v2f*)(p0 + 2 * NFACE * 4); // edge 2
#pragma unroll
        for (int t = 0; t < NP; ++t) {
            v8f z = {};
            v8f c0 = __builtin_amdgcn_wmma_f32_16x16x4_f32(
                false, a0, false, bfrag[t], (short)0, z, false, false);
            v8f c1 = __builtin_amdgcn_wmma_f32_16x16x4_f32(
                false, a1, false, bfrag[t], (short)0, z, false, false);
            v8f c2 = __builtin_amdgcn_wmma_f32_16x16x4_f32(
                false, a2, false, bfrag[t], (short)0, z, false, false);
            float lo[8], hi[8];
#pragma unroll
            for (int r = 0; r < 8; ++r) {
                lo[r] = fminf(fminf(c0[r], c1[r]), c2[r]);  // v_min3
                hi[r] = fmaxf(fmaxf(c0[r], c1[r]), c2[r]);  // v_max3
            }
            // reduce the 8 rows with v_max3 / v_min3 chains, folding the
            // loop-carried accumulator into the final 3-input op
            float p1 = fmaxf(fmaxf(lo[0], lo[1]), lo[2]);
            float p2 = fmaxf(fmaxf(p1, lo[3]), lo[4]);
            float p3 = fmaxf(fmaxf(p2, lo[5]), lo[6]);
            accP[t]  = fmaxf(fmaxf(p3, lo[7]), accP[t]);
            float n1 = fminf(fminf(hi[0], hi[1]), hi[2]);
            float n2 = fminf(fminf(n1, hi[3]), hi[4]);
            float n3 = fminf(fminf(n2, hi[5]), hi[6]);
            accN[t]  = fminf(fminf(n3, hi[7]), accN[t]);
        }
    }

    // ---- combine lane halves (rows M=0..7 in lanes 0-15, M=8..15 in 16-31)
    float* rrow = raster + (size_t)b * RSIZE * RSIZE + (size_t)ry * RSIZE + rx0;
#pragma unroll
    for (int t = 0; t < NP; ++t) {
        float mP = fmaxf(accP[t], __shfl_xor(accP[t], 16, 32));
        float mN = fminf(accN[t], __shfl_xor(accN[t], 16, 32));
        float cov = (mP > 0.0f || mN < 0.0f) ? 1.0f : 0.0f;
        if (lowh) rrow[t * 16 + fl] = cov;
    }
}

// ---------------------------------------------------------------------------
// Kernel 3: vertical flip + 2x2 mean pool -> out[b][128][128]
// ---------------------------------------------------------------------------
__global__ void rast_pool(const float* __restrict__ raster,
                          float* __restrict__ out) {
    int i = blockIdx.x * blockDim.x + threadIdx.x;   // [0, NB*128*128)
    if (i >= NB * 128 * 128) return;
    int b  = i >> 14;
    int r  = i & 16383;
    int oy = r >> 7;
    int ox = r & 127;
    const float* R = raster + (size_t)b * RSIZE * RSIZE;
    int r0 = (RSIZE - 1) - 2 * oy;   // flipped row pair
    int r1 = r0 - 1;
    int cx = 2 * ox;
    out[i] = 0.25f * (R[r0 * RSIZE + cx] + R[r0 * RSIZE + cx + 1] +
                      R[r1 * RSIZE + cx] + R[r1 * RSIZE + cx + 1]);
}

extern "C" void kernel_launch(void* const* d_in, const int* in_sizes, int n_in,
                              void* d_out, int out_size, void* d_ws, size_t ws_size,
                              hipStream_t stream) {
    const float* verts = (const float*)d_in[0];   // (4,1024,2) f32
    const int*   faces = (const int*)d_in[1];     // (4,1024,3) i32
    float*       out   = (float*)d_out;           // (4,128,128) f32

    // workspace: [0,192KB) = float4 coef[4][3][1024]; then raster 4*256*256 f32
    float4* coef   = (float4*)d_ws;
    float*  raster = (float*)((char*)d_ws + (size_t)NB * 3 * NFACE * sizeof(float4));

    rast_prep<<<(NB * 3 * NFACE + 255) / 256, 256, 0, stream>>>(verts, faces, coef);

    // 4096 waves total = NB * (65536 px / 64 px-per-wave); 8 waves per block
    rast_main<<<512, 256, 0, stream>>>((const float*)coef, raster);

    rast_pool<<<(NB * 128 * 128 + 255) / 256, 256, 0, stream>>>(raster, out);
}